// GINConvolution_6674379178025
// MI455X (gfx1250) — compile-verified
//
#include <hip/hip_runtime.h>

// GIN convolution for MI455X (gfx1250):
//   AX = S0@x + S1@x           (COO SpMM, f32 atomics, L2-resident x)
//   Z  = AX@W1 + b1            (wave WMMA f32 16x16x4, fused BN-stats)
//   h  = relu(BN(Z))           (folded into GEMM2 A-fragment build)
//   out= h@W2 + b2             (wave WMMA f32 16x16x4)

typedef float v2f __attribute__((ext_vector_type(2)));
typedef float v8f __attribute__((ext_vector_type(8)));

constexpr int NNODES = 100000;
constexpr int DIN    = 256;
constexpr int HID    = 64;
constexpr int OUTD   = 64;
constexpr int NEDGES = 3200000;
constexpr float BN_EPS = 1e-5f;
constexpr int ROW_TILES = NNODES / 16;   // 6250 (exact)

// ---------------------------------------------------------------------------
// SpMM: one wave per edge; lane l handles features l, l+32, ..., l+224 so
// every load / atomic instruction covers 128 contiguous bytes per wave.
// ---------------------------------------------------------------------------
__global__ __launch_bounds__(256) void spmm_atomic_kernel(
    const float* __restrict__ x, const int* __restrict__ rows,
    const int* __restrict__ cols, const float* __restrict__ vals,
    float* __restrict__ AX, int nEdges)
{
    const int gtid = blockIdx.x * blockDim.x + threadIdx.x;
    const int e    = gtid >> 5;
    const int lane = threadIdx.x & 31;
    if (e >= nEdges) return;

    const int   r = rows[e];
    const int   c = cols[e];
    const float v = vals[e];

    const float* __restrict__ xr = x  + (size_t)c * DIN;
    float*       __restrict__ ar = AX + (size_t)r * DIN;

#pragma unroll
    for (int j = 0; j < DIN / 32; ++j) {
        const int f = lane + 32 * j;
        atomicAdd(&ar[f], v * xr[f]);     // -> global_atomic_add_f32
    }
}

// ---------------------------------------------------------------------------
// GEMM1: Z = AX @ W1 + b1, with fused per-column sum / sum-of-squares for BN.
// One wave -> 16 rows x 64 cols via 4 accumulators of V_WMMA_F32_16X16X4_F32.
// ---------------------------------------------------------------------------
__global__ __launch_bounds__(256) void gemm1_bnstats_kernel(
    const float* __restrict__ AX, const float* __restrict__ W1,
    const float* __restrict__ b1, float* __restrict__ Z,
    float* __restrict__ colsum, float* __restrict__ colsumsq)
{
    __shared__ float ls [HID];
    __shared__ float ls2[HID];

    const int tid   = threadIdx.x;
    const int wave  = tid >> 5;
    const int lane  = tid & 31;
    const int halfk = lane >> 4;       // 0: K pair {k,k+1}, 1: K pair {k+2,k+3}
    const int m     = lane & 15;       // M for A / N for B,C,D

    const int rowTile = blockIdx.x * 8 + wave;
    const bool active = rowTile < ROW_TILES;   // wave-uniform -> EXEC all 1s
    const int row0    = rowTile * 16;

    if (tid < HID) { ls[tid] = 0.0f; ls2[tid] = 0.0f; }

    float s[4]  = {0.f, 0.f, 0.f, 0.f};
    float s2[4] = {0.f, 0.f, 0.f, 0.f};

    if (active) {
        v8f acc[4];
#pragma unroll
        for (int t = 0; t < 4; ++t) acc[t] = (v8f){0,0,0,0,0,0,0,0};

        const float* __restrict__ arow = AX + (size_t)(row0 + m) * DIN;

        for (int k0 = 0; k0 < DIN; k0 += 4) {
            const int kk = k0 + 2 * halfk;
            const v2f a = *(const v2f*)(arow + kk);          // A 16x4 fragment
            const float* __restrict__ w0 = W1 + (size_t)kk      * HID + m;
            const float* __restrict__ w1 = W1 + (size_t)(kk + 1) * HID + m;
#pragma unroll
            for (int t = 0; t < 4; ++t) {
                v2f b;                                        // B 4x16 fragment
                b.x = w0[t * 16];
                b.y = w1[t * 16];
                acc[t] = __builtin_amdgcn_wmma_f32_16x16x4_f32(
                    false, a, false, b, (short)0, acc[t], false, false);
            }
        }

        // epilogue: +b1, store Z, per-lane BN partials
#pragma unroll
        for (int t = 0; t < 4; ++t) {
            const int col   = t * 16 + m;
            const float bia = b1[col];
#pragma unroll
            for (int v = 0; v < 8; ++v) {
                const int r = row0 + v + 8 * halfk;           // C/D VGPR layout
                const float z = acc[t][v] + bia;
                Z[(size_t)r * HID + col] = z;
                s[t]  += z;
                s2[t] += z * z;
            }
        }
    }

    __syncthreads();
    if (active) {
#pragma unroll
        for (int t = 0; t < 4; ++t) {
            atomicAdd(&ls [t * 16 + m], s[t]);                // -> ds_add_f32
            atomicAdd(&ls2[t * 16 + m], s2[t]);
        }
    }
    __syncthreads();
    if (tid < HID) {
        atomicAdd(&colsum  [tid], ls [tid]);
        atomicAdd(&colsumsq[tid], ls2[tid]);
    }
}

// ---------------------------------------------------------------------------
// BN finalize: fold mean/var/gamma/beta into per-column scale & shift.
// ---------------------------------------------------------------------------
__global__ void bn_finalize_kernel(
    const float* __restrict__ colsum, const float* __restrict__ colsumsq,
    const float* __restrict__ gamma, const float* __restrict__ beta,
    float* __restrict__ scale, float* __restrict__ shift)
{
    const int c = threadIdx.x;
    if (c < HID) {
        const float inv_n = 1.0f / (float)NNODES;
        const float mean  = colsum[c] * inv_n;
        const float var   = colsumsq[c] * inv_n - mean * mean;
        const float rs    = rsqrtf(var + BN_EPS);
        const float sc    = gamma[c] * rs;
        scale[c] = sc;
        shift[c] = beta[c] - mean * sc;
    }
}

// ---------------------------------------------------------------------------
// GEMM2: out = relu(Z*scale + shift) @ W2 + b2 (BN+ReLU fused into A frags).
// ---------------------------------------------------------------------------
__global__ __launch_bounds__(256) void gemm2_kernel(
    const float* __restrict__ Z, const float* __restrict__ W2,
    const float* __restrict__ b2, const float* __restrict__ scale,
    const float* __restrict__ shift, float* __restrict__ out)
{
    const int tid   = threadIdx.x;
    const int wave  = tid >> 5;
    const int lane  = tid & 31;
    const int halfk = lane >> 4;
    const int m     = lane & 15;

    const int rowTile = blockIdx.x * 8 + wave;
    if (rowTile >= ROW_TILES) return;                         // wave-uniform
    const int row0 = rowTile * 16;

    v8f acc[4];
#pragma unroll
    for (int t = 0; t < 4; ++t) acc[t] = (v8f){0,0,0,0,0,0,0,0};

    const float* __restrict__ zrow = Z + (size_t)(row0 + m) * HID;

    for (int k0 = 0; k0 < HID; k0 += 4) {
        const int kk = k0 + 2 * halfk;
        const v2f zv = *(const v2f*)(zrow + kk);
        v2f a;                                                // BN + ReLU fused
        a.x = fmaxf(zv.x * scale[kk]     + shift[kk],     0.0f);
        a.y = fmaxf(zv.y * scale[kk + 1] + shift[kk + 1], 0.0f);
        const float* __restrict__ w0 = W2 + (size_t)kk      * OUTD + m;
        const float* __restrict__ w1 = W2 + (size_t)(kk + 1) * OUTD + m;
#pragma unroll
        for (int t = 0; t < 4; ++t) {
            v2f b;
            b.x = w0[t * 16];
            b.y = w1[t * 16];
            acc[t] = __builtin_amdgcn_wmma_f32_16x16x4_f32(
                false, a, false, b, (short)0, acc[t], false, false);
        }
    }

#pragma unroll
    for (int t = 0; t < 4; ++t) {
        const int col   = t * 16 + m;
        const float bia = b2[col];
#pragma unroll
        for (int v = 0; v < 8; ++v) {
            const int r = row0 + v + 8 * halfk;
            out[(size_t)r * OUTD + col] = acc[t][v] + bia;
        }
    }
}

// ---------------------------------------------------------------------------
extern "C" void kernel_launch(void* const* d_in, const int* in_sizes, int n_in,
                              void* d_out, int out_size, void* d_ws, size_t ws_size,
                              hipStream_t stream)
{
    (void)in_sizes; (void)n_in; (void)out_size; (void)ws_size;

    const float* x     = (const float*)d_in[0];
    const int*   rows0 = (const int*)  d_in[1];
    const int*   cols0 = (const int*)  d_in[2];
    const float* vals0 = (const float*)d_in[3];
    const int*   rows1 = (const int*)  d_in[4];
    const int*   cols1 = (const int*)  d_in[5];
    const float* vals1 = (const float*)d_in[6];
    const float* W1    = (const float*)d_in[7];
    const float* b1    = (const float*)d_in[8];
    const float* gamma = (const float*)d_in[9];
    const float* beta  = (const float*)d_in[10];
    const float* W2    = (const float*)d_in[11];
    const float* b2    = (const float*)d_in[12];
    float*       out   = (float*)d_out;

    // workspace layout
    float* AX       = (float*)d_ws;                        // 100000*256
    float* Z        = AX + (size_t)NNODES * DIN;           // 100000*64
    float* colsum   = Z  + (size_t)NNODES * HID;           // 64
    float* colsumsq = colsum + HID;                        // 64
    float* scale    = colsumsq + HID;                      // 64
    float* shift    = scale + HID;                         // 64

    // zero the atomic accumulators (graph-capturable)
    hipMemsetAsync(AX, 0, (size_t)NNODES * DIN * sizeof(float), stream);
    hipMemsetAsync(colsum, 0, 2 * HID * sizeof(float), stream);

    // SpMM over both supports (8 edges per 256-thread block, 1 edge per wave)
    const int spmmBlocks = (NEDGES + 7) / 8;
    spmm_atomic_kernel<<<spmmBlocks, 256, 0, stream>>>(x, rows0, cols0, vals0, AX, NEDGES);
    spmm_atomic_kernel<<<spmmBlocks, 256, 0, stream>>>(x, rows1, cols1, vals1, AX, NEDGES);

    const int gemmBlocks = (ROW_TILES + 7) / 8;            // 782
    gemm1_bnstats_kernel<<<gemmBlocks, 256, 0, stream>>>(AX, W1, b1, Z, colsum, colsumsq);
    bn_finalize_kernel<<<1, 64, 0, stream>>>(colsum, colsumsq, gamma, beta, scale, shift);
    gemm2_kernel<<<gemmBlocks, 256, 0, stream>>>(Z, W2, b2, scale, shift, out);
}